// GraphQNNHybrid_65481071403791
// MI455X (gfx1250) — compile-verified
//
#include <hip/hip_runtime.h>
#include <math.h>

typedef __attribute__((ext_vector_type(2))) float v2f;
typedef __attribute__((ext_vector_type(8))) float v8f;

#define NDIM 8192
#define KSPLIT 4
#define KSEG (NDIM / KSPLIT)                       // 2048 columns per block
#define ROWS_PER_WAVE 16
#define WAVES_PER_BLOCK 8
#define ROWS_PER_BLOCK (ROWS_PER_WAVE * WAVES_PER_BLOCK)  // 128

// ---------------------------------------------------------------------------
// Stage 1: per-node "conv" = dot(features[16], kernel[16]) + b, then sigmoid.
// ---------------------------------------------------------------------------
__global__ void conv_sigmoid_kernel(const float* __restrict__ nf,
                                    const float* __restrict__ cw,
                                    const float* __restrict__ cb,
                                    float* __restrict__ xout) {
    const int i = blockIdx.x * blockDim.x + threadIdx.x;
    const float4* f = (const float4*)(nf + (size_t)i * 16);
    const float4* w = (const float4*)cw;
    float s = 0.0f;
#pragma unroll
    for (int j = 0; j < 4; ++j) {
        float4 a = f[j];
        float4 b = w[j];
        s += a.x * b.x + a.y * b.y + a.z * b.z + a.w * b.w;
    }
    s += cb[0];
    xout[i] = 1.0f / (1.0f + expf(-s));
}

// ---------------------------------------------------------------------------
// Stage 2/4: GEMV via V_WMMA_F32_16X16X4_F32.
// Wave = 16 rows x KSEG cols. A lane layout (ISA 7.12.2, 32-bit A 16x4):
//   lanes 0-15  -> logical K0,K1 ; lanes 16-31 -> logical K2,K3.
// We remap physical k so each lane's float4 load feeds two WMMAs:
//   wmma#1 consumes {.x,.y} of lo lanes (k+0,1) and hi lanes (k+4,5)
//   wmma#2 consumes {.z,.w} of lo lanes (k+2,3) and hi lanes (k+6,7)
// B is the x-vector replicated across all 16 columns, read from LDS with the
// identical lane layout, so every accumulator column holds the GEMV result.
// ---------------------------------------------------------------------------
__global__ __launch_bounds__(256) void gemv_wmma_kernel(
        const float* __restrict__ W,
        const float* __restrict__ x,
        float* __restrict__ part) {
    __shared__ float xs[KSEG];

    const int tid   = threadIdx.x;
    const int kbase = blockIdx.y * KSEG;

    // Coalesced preload of the x segment into LDS (8 KiB).
    {
        const float4* xg = (const float4*)(x + kbase);
        float4* xl = (float4*)xs;
        for (int i = tid; i < KSEG / 4; i += blockDim.x) xl[i] = xg[i];
    }
    __syncthreads();

    const int wave = tid >> 5;
    const int lane = tid & 31;
    const int hi   = lane >> 4;                    // 0: lanes 0-15, 1: lanes 16-31
    const int r0   = blockIdx.x * ROWS_PER_BLOCK + wave * ROWS_PER_WAVE;
    const int row  = r0 + (lane & 15);

    const float4* Ap = (const float4*)(W + (size_t)row * NDIM + kbase) + hi;
    const float4* Bp = (const float4*)xs + hi;

    v8f acc0 = {};
    v8f acc1 = {};

#pragma unroll 4
    for (int t = 0; t < KSEG / 8; ++t) {
        float4 a  = Ap[2 * t];                     // global_load_b128, streaming W
        float4 xv = Bp[2 * t];                     // ds_read_b128, x segment
        v2f a01 = {a.x, a.y};
        v2f a23 = {a.z, a.w};
        v2f b01 = {xv.x, xv.y};
        v2f b23 = {xv.z, xv.w};
        acc0 = __builtin_amdgcn_wmma_f32_16x16x4_f32(
                   false, a01, false, b01, (short)0, acc0, false, false);
        acc1 = __builtin_amdgcn_wmma_f32_16x16x4_f32(
                   false, a23, false, b23, (short)0, acc1, false, false);
    }

    v8f acc = acc0 + acc1;

    // C/D layout: VGPR j = row j (lanes 0-15) or row 8+j (lanes 16-31).
    // Lane 0 writes rows r0..r0+7, lane 16 writes rows r0+8..r0+15.
    if ((lane & 15) == 0) {
        float* dst = part + (size_t)blockIdx.y * NDIM + r0 + hi * 8;
#pragma unroll
        for (int j = 0; j < 8; ++j) dst[j] = acc[j];
    }
}

// ---------------------------------------------------------------------------
// Stage 3/5: sum K-split partials + bias, optional tanh. Deterministic.
// ---------------------------------------------------------------------------
__global__ void reduce_kernel(const float* __restrict__ part,
                              const float* __restrict__ bias,
                              float* __restrict__ out,
                              int do_tanh) {
    const int i = blockIdx.x * blockDim.x + threadIdx.x;
    float s = bias[i];
#pragma unroll
    for (int ks = 0; ks < KSPLIT; ++ks) s += part[(size_t)ks * NDIM + i];
    out[i] = do_tanh ? tanhf(s) : s;
}

extern "C" void kernel_launch(void* const* d_in, const int* in_sizes, int n_in,
                              void* d_out, int out_size, void* d_ws, size_t ws_size,
                              hipStream_t stream) {
    (void)in_sizes; (void)n_in; (void)out_size; (void)ws_size;

    const float* nf = (const float*)d_in[0];   // node_features (8192,16)
    const float* cw = (const float*)d_in[1];   // conv_w (16,)
    const float* cb = (const float*)d_in[2];   // conv_b scalar
    const float* W1 = (const float*)d_in[3];   // (8192,8192)
    const float* b1 = (const float*)d_in[4];   // (8192,)
    const float* W2 = (const float*)d_in[5];   // (8192,8192)
    const float* b2 = (const float*)d_in[6];   // (8192,)
    float* out = (float*)d_out;                // (8192,)

    float* ws = (float*)d_ws;
    float* x  = ws;                            // 8192 floats
    float* h  = ws + NDIM;                     // 8192 floats
    float* p  = ws + 2 * NDIM;                 // 8192*KSPLIT floats (reused)

    conv_sigmoid_kernel<<<NDIM / 256, 256, 0, stream>>>(nf, cw, cb, x);

    dim3 g(NDIM / ROWS_PER_BLOCK, KSPLIT);     // 64 x 4 blocks, 8 waves each
    gemv_wmma_kernel<<<g, 256, 0, stream>>>(W1, x, p);
    reduce_kernel<<<NDIM / 256, 256, 0, stream>>>(p, b1, h, 1);
    gemv_wmma_kernel<<<g, 256, 0, stream>>>(W2, h, p);
    reduce_kernel<<<NDIM / 256, 256, 0, stream>>>(p, b2, out, 0);
}